// GCN_2190433321521
// MI455X (gfx1250) — compile-verified
//
#include <hip/hip_runtime.h>

typedef float v2f __attribute__((ext_vector_type(2)));
typedef float v8f __attribute__((ext_vector_type(8)));
typedef unsigned int v4u __attribute__((ext_vector_type(4)));
typedef int v4i __attribute__((ext_vector_type(4)));
typedef int v8i __attribute__((ext_vector_type(8)));

#define NFEAT 128
#define NCLS  40
#define S1    72    // LDS row stride (floats) for GEMM1 slice: 64 + 8 pad -> conflict-free
#define S2    48    // LDS row stride for GEMM2: 40 padded to 48 -> conflict-free

#if __has_builtin(__builtin_amdgcn_tensor_load_to_lds) && __has_builtin(__builtin_amdgcn_s_wait_tensorcnt)
#define USE_TDM 1
#else
#define USE_TDM 0
#endif

#if USE_TDM
// ---------------------------------------------------------------------------
// Tensor Data Mover: 2D tile Global->LDS per CDNA5 D# descriptor (ISA 08 §8).
//   g0: [1:0]=count=1, [63:32]=lds_addr, [120:64]=global_addr, [127:126]=type=2
//   g1: [17:16]=data_size(2=4B), [20]=pad_en, [24:22]=pad_interval, [31:25]=pad_amt,
//       [79:48]=tensor_dim0, [111:80]=tensor_dim1, [127:112]=tile_dim0,
//       [143:128]=tile_dim1, [207:160]=tensor_dim0_stride
// All operands wave-uniform -> SGPR descriptor groups.
// This toolchain exposes the 6-arg builtin: (v4u, v8i, v4i, v4i, v8i, i32 cpol).
// ---------------------------------------------------------------------------
__device__ __forceinline__
void tdm_load_2d(const void* gaddr, unsigned lds_addr,
                 unsigned dim0, unsigned dim1, unsigned tile0, unsigned tile1,
                 unsigned stride0, unsigned padbits) {
  unsigned long long ga = (unsigned long long)gaddr;
  v4u g0;
  g0[0] = 1u;                                          // count=1, user mode
  g0[1] = lds_addr;                                    // LDS byte address
  g0[2] = (unsigned)ga;                                // global_addr[31:0]
  g0[3] = (unsigned)((ga >> 32) & 0x1FFFFFFull)        // global_addr[56:32]
          | (2u << 30);                                // type=2 ("image")
  v8i g1;
  g1[0] = (int)((2u << 16) | padbits);                 // data_size=4B (+pad ctl)
  g1[1] = (int)((dim0 & 0xFFFFu) << 16);               // tensor_dim0[15:0]
  g1[2] = (int)((dim0 >> 16) | ((dim1 & 0xFFFFu) << 16));
  g1[3] = (int)((dim1 >> 16) | (tile0 << 16));         // tile_dim0
  g1[4] = (int)(tile1 & 0xFFFFu);                      // tile_dim1 (tile_dim2=0)
  g1[5] = (int)stride0;                                // tensor_dim0_stride[31:0]
  g1[6] = 0;                                           // stride0 hi / dim1_stride
  g1[7] = 0;
  v4i z4 = {0, 0, 0, 0};                               // groups 2/3: unused (2D)
  v8i z8 = {0, 0, 0, 0, 0, 0, 0, 0};                   // extra arg (clang-23 form)
  __builtin_amdgcn_tensor_load_to_lds(g0, g1, z4, z4, z8, 0);
}
#endif

// ---------------------------------------------------------------------------
// 1) Segment-wise input projection: x[n,f] = b[f] + sum_k in[n,k] * W[k,f]
// ---------------------------------------------------------------------------
__global__ void proj_kernel(const float* __restrict__ xr, const float* __restrict__ xu,
                            const float* __restrict__ xp,
                            const float* __restrict__ Wr, const float* __restrict__ br,
                            const float* __restrict__ Wu, const float* __restrict__ bu,
                            const float* __restrict__ Wp, const float* __restrict__ bp,
                            float* __restrict__ out, int NR, int NU, int NP) {
  int tid = blockIdx.x * blockDim.x + threadIdx.x;
  int n = tid >> 7;
  int f = tid & 127;
  int Ntot = NR + NU + NP;
  if (n >= Ntot) return;
  const float* in; const float* W; const float* b; int K; int nn;
  if (n < NR)            { in = xr; W = Wr; b = br; K = 5; nn = n; }
  else if (n < NR + NU)  { in = xu; W = Wu; b = bu; K = 7; nn = n - NR; }
  else                   { in = xp; W = Wp; b = bp; K = 6; nn = n - NR - NU; }
  const float* row = in + (size_t)nn * K;
  float acc = b[f];
  #pragma unroll 1
  for (int k = 0; k < K; ++k) acc = fmaf(row[k], W[k * NFEAT + f], acc);
  out[(size_t)n * NFEAT + f] = acc;
}

// ---------------------------------------------------------------------------
// 2) GEMM1: C[N,128] = A[N,128] @ W[128,128] via V_WMMA_F32_16X16X4_F32.
//    W slice (128 x 64, row-major, LDS stride 72) staged by TDM.
// ---------------------------------------------------------------------------
__global__ __launch_bounds__(256)
void gemm1_kernel(const float* __restrict__ A, const float* __restrict__ W,
                  float* __restrict__ C, int Nrows) {
  __shared__ float Wl[128 * S1];             // 36.9 KB
  const int tid = threadIdx.x;
  const int nhalf = blockIdx.y;              // column half: [nhalf*64, +64)

#if USE_TDM
  if (tid < 32) {
    // pad_interval: 64 dwords (enc 5), pad_amount: 8 dwords (enc 7) -> stride 72
    tdm_load_2d(W + nhalf * 64, (unsigned)(unsigned long long)&Wl[0],
                /*dim0*/64, /*dim1*/128, /*tile0*/64, /*tile1*/128,
                /*stride0*/NFEAT, (1u << 20) | (5u << 22) | (7u << 25));
    __builtin_amdgcn_s_wait_tensorcnt(0);
  }
#else
  for (int idx = tid; idx < 128 * 64; idx += 256) {
    int k = idx >> 6, n = idx & 63;
    Wl[k * S1 + n] = W[k * NFEAT + nhalf * 64 + n];
  }
#endif
  __syncthreads();

  const int wave = tid >> 5;
  const int lane = tid & 31;
  const int tile = blockIdx.x * 8 + wave;
  const int mbase = tile * 16;
  if (mbase >= Nrows) return;

  const int l15   = lane & 15;
  const int khalf = lane >> 4;               // A frag: K pair {0,1} or {2,3}
  v8f acc0 = {}, acc1 = {}, acc2 = {}, acc3 = {};

  int row = mbase + l15; if (row >= Nrows) row = Nrows - 1;
  const float* arow = A + (size_t)row * NFEAT + (khalf << 1);
  const float* bb   = Wl + (khalf << 1) * S1 + l15;   // rows kb+2*khalf, cols l15

  #pragma unroll 8
  for (int kb = 0; kb < NFEAT; kb += 4) {
    v2f a = *(const v2f*)(arow + kb);
    v2f b0, b1, b2, b3;
    const float* p = bb + kb * S1;
    b0.x = p[ 0];      b0.y = p[S1 +  0];
    b1.x = p[16];      b1.y = p[S1 + 16];
    b2.x = p[32];      b2.y = p[S1 + 32];
    b3.x = p[48];      b3.y = p[S1 + 48];
    acc0 = __builtin_amdgcn_wmma_f32_16x16x4_f32(false, a, false, b0, (short)0, acc0, false, false);
    acc1 = __builtin_amdgcn_wmma_f32_16x16x4_f32(false, a, false, b1, (short)0, acc1, false, false);
    acc2 = __builtin_amdgcn_wmma_f32_16x16x4_f32(false, a, false, b2, (short)0, acc2, false, false);
    acc3 = __builtin_amdgcn_wmma_f32_16x16x4_f32(false, a, false, b3, (short)0, acc3, false, false);
  }

  const int colbase = nhalf * 64 + l15;
  #pragma unroll
  for (int r = 0; r < 8; ++r) {
    int m = mbase + r + (khalf << 3);        // C: VGPR r -> M = r + 8*(lane>=16)
    if (m < Nrows) {
      float* crow = C + (size_t)m * NFEAT + colbase;
      crow[0]  = acc0[r];
      crow[16] = acc1[r];
      crow[32] = acc2[r];
      crow[48] = acc3[r];
    }
  }
}

// ---------------------------------------------------------------------------
// 3) GEMM2: C[N,40] = relu(A)[N,128] @ W[128,40]; W staged by TDM with
//    tile_dim0=48 > tensor_dim0=40 (TDM OOB zero-fill pads cols 40..47).
// ---------------------------------------------------------------------------
__global__ __launch_bounds__(256)
void gemm2_kernel(const float* __restrict__ A, const float* __restrict__ W,
                  float* __restrict__ C, int Nrows) {
  __shared__ float Wl[128 * S2];             // 24.6 KB
  const int tid = threadIdx.x;

#if USE_TDM
  if (tid < 32) {
    tdm_load_2d(W, (unsigned)(unsigned long long)&Wl[0],
                /*dim0*/NCLS, /*dim1*/128, /*tile0*/S2, /*tile1*/128,
                /*stride0*/NCLS, 0u);
    __builtin_amdgcn_s_wait_tensorcnt(0);
  }
#else
  for (int idx = tid; idx < 128 * S2; idx += 256) {
    int k = idx / S2, n = idx - k * S2;
    Wl[idx] = (n < NCLS) ? W[k * NCLS + n] : 0.0f;
  }
#endif
  __syncthreads();

  const int wave = tid >> 5;
  const int lane = tid & 31;
  const int tile = blockIdx.x * 8 + wave;
  const int mbase = tile * 16;
  if (mbase >= Nrows) return;

  const int l15   = lane & 15;
  const int khalf = lane >> 4;
  v8f acc0 = {}, acc1 = {}, acc2 = {};

  int row = mbase + l15; if (row >= Nrows) row = Nrows - 1;
  const float* arow = A + (size_t)row * NFEAT + (khalf << 1);
  const float* bb   = Wl + (khalf << 1) * S2 + l15;

  #pragma unroll 8
  for (int kb = 0; kb < NFEAT; kb += 4) {
    v2f a = *(const v2f*)(arow + kb);
    a.x = fmaxf(a.x, 0.0f);                  // fused relu on aggregated features
    a.y = fmaxf(a.y, 0.0f);
    v2f b0, b1, b2;
    const float* p = bb + kb * S2;
    b0.x = p[ 0];      b0.y = p[S2 +  0];
    b1.x = p[16];      b1.y = p[S2 + 16];
    b2.x = p[32];      b2.y = p[S2 + 32];
    acc0 = __builtin_amdgcn_wmma_f32_16x16x4_f32(false, a, false, b0, (short)0, acc0, false, false);
    acc1 = __builtin_amdgcn_wmma_f32_16x16x4_f32(false, a, false, b1, (short)0, acc1, false, false);
    acc2 = __builtin_amdgcn_wmma_f32_16x16x4_f32(false, a, false, b2, (short)0, acc2, false, false);
  }

  #pragma unroll
  for (int r = 0; r < 8; ++r) {
    int m = mbase + r + (khalf << 3);
    if (m < Nrows) {
      float* crow = C + (size_t)m * NCLS;
      crow[l15]      = acc0[r];
      crow[16 + l15] = acc1[r];
      if (32 + l15 < NCLS) crow[32 + l15] = acc2[r];
    }
  }
}

// ---------------------------------------------------------------------------
// 4) Edge scatter: out[dst[e], :] += feat[src[e], :]  (wave32 per edge,
//    coalesced gathers, native f32 global atomics; resolves in 192MB L2)
// ---------------------------------------------------------------------------
__global__ void scatter_kernel(const float* __restrict__ feat,
                               const int* __restrict__ src, const int* __restrict__ dst,
                               float* __restrict__ out, int E, int F) {
  int tid = blockIdx.x * blockDim.x + threadIdx.x;
  int e = tid >> 5;
  int lane = tid & 31;
  if (e >= E) return;
  int s = src[e];
  int d = dst[e];
  const float* sp = feat + (size_t)s * F;
  float* dp = out + (size_t)d * F;
  for (int f = lane; f < F; f += 32)
    unsafeAtomicAdd(dp + f, sp[f]);
}

// ---------------------------------------------------------------------------
extern "C" void kernel_launch(void* const* d_in, const int* in_sizes, int n_in,
                              void* d_out, int out_size, void* d_ws, size_t ws_size,
                              hipStream_t stream) {
  const float* xr = (const float*)d_in[0];
  const float* xu = (const float*)d_in[1];
  const float* xp = (const float*)d_in[2];
  const int*   ei = (const int*)d_in[3];
  const float* Wr = (const float*)d_in[4];
  const float* br = (const float*)d_in[5];
  const float* Wu = (const float*)d_in[6];
  const float* bu = (const float*)d_in[7];
  const float* Wp = (const float*)d_in[8];
  const float* bp = (const float*)d_in[9];
  const float* W1 = (const float*)d_in[10];
  const float* W2 = (const float*)d_in[11];

  const int NR = in_sizes[0] / 5;
  const int NU = in_sizes[1] / 7;
  const int NP = in_sizes[2] / 6;
  const int N  = NR + NU + NP;
  const int E  = in_sizes[3] / 2;
  const int* src = ei;
  const int* dst = ei + E;

  float* bufA = (float*)d_ws;                     // x [N,128], later agg1
  float* bufB = bufA + (size_t)N * NFEAT;         // t1 = x@W1 [N,128]
  float* bufC = bufB + (size_t)N * NFEAT;         // t2 = relu(agg1)@W2 [N,40]
  float* out  = (float*)d_out;

  // 1) input projections -> bufA
  {
    long long total = (long long)N * NFEAT;
    int blocks = (int)((total + 255) / 256);
    proj_kernel<<<blocks, 256, 0, stream>>>(xr, xu, xp, Wr, br, Wu, bu, Wp, bp,
                                            bufA, NR, NU, NP);
  }
  // 2) t1 = x @ W1 -> bufB
  {
    int mtiles = (N + 15) / 16;
    dim3 grid((mtiles + 7) / 8, 2);
    gemm1_kernel<<<grid, 256, 0, stream>>>(bufA, W1, bufB, N);
  }
  // 3) agg1 = 0; agg1[dst] += t1[src]   (bufA reused as accumulator)
  (void)hipMemsetAsync(bufA, 0, (size_t)N * NFEAT * sizeof(float), stream);
  {
    long long th = (long long)E * 32;
    int blocks = (int)((th + 255) / 256);
    scatter_kernel<<<blocks, 256, 0, stream>>>(bufB, src, dst, bufA, E, NFEAT);
  }
  // 4) t2 = relu(agg1) @ W2 -> bufC
  {
    int mtiles = (N + 15) / 16;
    dim3 grid((mtiles + 7) / 8, 1);
    gemm2_kernel<<<grid, 256, 0, stream>>>(bufA, W2, bufC, N);
  }
  // 5) out = 0; out[dst] += t2[src]
  (void)hipMemsetAsync(d_out, 0, (size_t)N * NCLS * sizeof(float), stream);
  {
    long long th = (long long)E * 32;
    int blocks = (int)((th + 255) / 256);
    scatter_kernel<<<blocks, 256, 0, stream>>>(bufC, src, dst, out, E, NCLS);
  }
}